// RgcnMetaModelKai2Mod1Ly3_12833362280972
// MI455X (gfx1250) — compile-verified
//
#include <hip/hip_runtime.h>

typedef __attribute__((ext_vector_type(2))) float v2f;
typedef __attribute__((ext_vector_type(8))) float v8f;

#define N_NODES  51200
#define N_REL    4
#define D_IN     768
#define D_H      512
#define D_OUT    768
#define N_GRAPHS 50
#define N_PER    1024
#define N_FRONT  512

__global__ void zero_i32_kernel(int* __restrict__ p, int n) {
  int i = blockIdx.x * blockDim.x + threadIdx.x;
  if (i < n) p[i] = 0;
}

__global__ void count_edges_kernel(const int* __restrict__ dst, const int* __restrict__ et,
                                   int* __restrict__ cnt, int E) {
  int e = blockIdx.x * blockDim.x + threadIdx.x;
  if (e < E) atomicAdd(&cnt[et[e] * N_NODES + dst[e]], 1);
}

// Single-block exclusive scan over n ints (n ~ 204800): writes rowstart and cursor.
__global__ void exscan_kernel(const int* __restrict__ in, int* __restrict__ rowstart,
                              int* __restrict__ cursor, int n) {
  __shared__ int smem[1024];
  __shared__ int carry_s;
  int tid = threadIdx.x;
  if (tid == 0) carry_s = 0;
  __syncthreads();
  for (int base = 0; base < n; base += 1024) {
    int v = (base + tid < n) ? in[base + tid] : 0;
    smem[tid] = v;
    __syncthreads();
    for (int d = 1; d < 1024; d <<= 1) {       // Hillis-Steele inclusive scan
      int t = (tid >= d) ? smem[tid - d] : 0;
      __syncthreads();
      smem[tid] += t;
      __syncthreads();
    }
    int carry = carry_s;
    int excl = carry + smem[tid] - v;
    if (base + tid < n) { rowstart[base + tid] = excl; cursor[base + tid] = excl; }
    __syncthreads();
    if (tid == 0) carry_s = carry + smem[1023];
    __syncthreads();
  }
}

// Bucket each edge's source node into the (relation,dst) CSR slot.
__global__ void fill_srclist_kernel(const int* __restrict__ src, const int* __restrict__ dst,
                                    const int* __restrict__ et, int* __restrict__ cursor,
                                    int* __restrict__ srclist, int E) {
  int e = blockIdx.x * blockDim.x + threadIdx.x;
  if (e >= E) return;
  int pos = atomicAdd(&cursor[et[e] * N_NODES + dst[e]], 1);
  srclist[pos] = src[e];
}

// C[M x Dout] = A[M x K] @ W[K x Dout] (+ bias). One wave -> 64x64 output tile
// (4x4 grid of 16x16 fp32 WMMA accumulators, ~190 VGPRs -> need relaxed occupancy cap).
__global__ void __launch_bounds__(256, 2)
gemm_wmma_f32_kernel(const float* __restrict__ A, const float* __restrict__ W,
                     const float* __restrict__ bias, float* __restrict__ C,
                     int M, int K, int Dout, int has_bias) {
  int wave = (int)((blockIdx.x * (unsigned)blockDim.x + threadIdx.x) >> 5);
  int lane = threadIdx.x & 31;
  int ntile = Dout >> 6;                 // 64-column tiles
  int row0 = (wave / ntile) << 6;        // 64-row tiles
  int col0 = (wave % ntile) << 6;
  if (row0 >= M) return;                 // wave-uniform
  int m  = lane & 15;
  int hi = lane >> 4;

  v8f zero = {};
  v8f acc[4][4];
  #pragma unroll
  for (int i = 0; i < 4; ++i)
    #pragma unroll
    for (int t = 0; t < 4; ++t)
      acc[i][t] = zero;

  // A 16x4 fragment layout: lanes 0-15 hold K=k..k+1, lanes 16-31 hold K=k+2..k+3
  const float* ap0 = A + (size_t)(row0 + m) * K + 2 * hi;
  const float* ap1 = ap0 + (size_t)16 * K;
  const float* ap2 = ap1 + (size_t)16 * K;
  const float* ap3 = ap2 + (size_t)16 * K;
  // B 4x16 fragment: VGPR j holds row (k + 2*hi + j), column col0 + (lane&15)
  const float* wp = W + (size_t)(2 * hi) * Dout + col0 + m;
  const size_t wstep = (size_t)4 * Dout;

  for (int k = 0; k < K; k += 4) {
    v2f a[4];
    a[0] = *(const v2f*)(ap0 + k);
    a[1] = *(const v2f*)(ap1 + k);
    a[2] = *(const v2f*)(ap2 + k);
    a[3] = *(const v2f*)(ap3 + k);
    const float* w0 = wp;
    const float* w1 = wp + Dout;
    v2f b[4];
    #pragma unroll
    for (int t = 0; t < 4; ++t) {
      v2f bt = { w0[16 * t], w1[16 * t] };
      b[t] = bt;
    }
    wp += wstep;
    #pragma unroll
    for (int i = 0; i < 4; ++i)
      #pragma unroll
      for (int t = 0; t < 4; ++t)
        acc[i][t] = __builtin_amdgcn_wmma_f32_16x16x4_f32(
            false, a[i], false, b[t], (short)0, acc[i][t], false, false);
  }

  // C/D layout: VGPR v, lane l -> row (v + 8*hi), col (l&15)
  #pragma unroll
  for (int i = 0; i < 4; ++i) {
    #pragma unroll
    for (int t = 0; t < 4; ++t) {
      int col = col0 + 16 * t + m;
      float bv = has_bias ? bias[col] : 0.0f;
      float* cp = C + (size_t)(row0 + 16 * i + 8 * hi) * Dout + col;
      #pragma unroll
      for (int v = 0; v < 8; ++v)
        cp[(size_t)v * Dout] = acc[i][t][v] + bv;
    }
  }
}

// acc[i] += mean over in-neighbors (relation r) of h[src]. One block per node; no atomics.
__global__ void gather_mean_kernel(const float* __restrict__ h, const int* __restrict__ rowstart,
                                   const int* __restrict__ cnt, const int* __restrict__ srclist,
                                   float* __restrict__ acc, int D, int r) {
  int i = blockIdx.x;                    // destination node
  int c = cnt[r * N_NODES + i];
  if (c == 0) return;                    // block-uniform
  int s0 = rowstart[r * N_NODES + i];
  float inv = 1.0f / (float)c;
  float* ai = acc + (size_t)i * D;
  for (int f = threadIdx.x; f < D; f += blockDim.x) {
    float sum = 0.0f;
    for (int j = 0; j < c; ++j) {
      int s = srclist[s0 + j];
      sum += h[(size_t)s * D + f];
    }
    ai[f] += sum * inv;
  }
}

__global__ void relu_kernel(float* __restrict__ p, long n) {
  long i = blockIdx.x * (long)blockDim.x + threadIdx.x;
  if (i < n) p[i] = fmaxf(p[i], 0.0f);
}

__global__ void pool_kernel(const float* __restrict__ h2, float* __restrict__ pooled) {
  int idx = blockIdx.x * blockDim.x + threadIdx.x;      // N_GRAPHS * 2 * D_OUT
  if (idx >= N_GRAPHS * 2 * D_OUT) return;
  int g = idx / (2 * D_OUT);
  int f = idx % (2 * D_OUT);
  int half = f / D_OUT;
  int feat = f % D_OUT;
  const float* base = h2 + ((size_t)g * N_PER + half * N_FRONT) * D_OUT + feat;
  float s = 0.0f;
  for (int i = 0; i < N_FRONT; ++i) s += base[(size_t)i * D_OUT];
  pooled[idx] = s * (1.0f / (float)N_FRONT);
}

__global__ void fc1_kernel(const float* __restrict__ pooled, const float* __restrict__ w,
                           const float* __restrict__ b, float* __restrict__ out) {
  int idx = blockIdx.x * blockDim.x + threadIdx.x;      // N_GRAPHS * 128
  if (idx >= N_GRAPHS * 128) return;
  int g = idx / 128, c = idx % 128;
  float s = b[c];
  const float* p = pooled + (size_t)g * (2 * D_OUT);
  for (int k = 0; k < 2 * D_OUT; ++k) s += p[k] * w[(size_t)k * 128 + c];
  out[idx] = fmaxf(s, 0.0f);
}

__global__ void fc2_kernel(const float* __restrict__ h, const float* __restrict__ w,
                           const float* __restrict__ b, float* __restrict__ out) {
  int idx = blockIdx.x * blockDim.x + threadIdx.x;      // N_GRAPHS * 2
  if (idx >= N_GRAPHS * 2) return;
  int g = idx / 2, c = idx % 2;
  float s = b[c];
  const float* p = h + (size_t)g * 128;
  for (int k = 0; k < 128; ++k) s += p[k] * w[k * 2 + c];
  out[idx] = s;
}

static inline int gemm_blocks(int M, int Dout) {
  // waves = (M/64)*(Dout/64), 8 waves (256 threads) per block
  return (M / 64) * (Dout / 64) / 8;
}

extern "C" void kernel_launch(void* const* d_in, const int* in_sizes, int n_in,
                              void* d_out, int out_size, void* d_ws, size_t ws_size,
                              hipStream_t stream) {
  const float* x       = (const float*)d_in[0];
  const int*   src     = (const int*)d_in[1];
  const int*   dst     = (const int*)d_in[2];
  const int*   etype   = (const int*)d_in[3];
  const float* W1_rel  = (const float*)d_in[4];
  const float* W1_root = (const float*)d_in[5];
  const float* b1      = (const float*)d_in[6];
  const float* W2_rel  = (const float*)d_in[7];
  const float* W2_root = (const float*)d_in[8];
  const float* b2      = (const float*)d_in[9];
  const float* fc1_w   = (const float*)d_in[10];
  const float* fc1_b   = (const float*)d_in[11];
  const float* fc2_w   = (const float*)d_in[12];
  const float* fc2_b   = (const float*)d_in[13];
  float* out = (float*)d_out;
  const int E = in_sizes[1];

  // Workspace layout (all chunks 256B-aligned)
  char* ws = (char*)d_ws;
  size_t off = 0;
  auto take = [&](size_t bytes) { void* p = ws + off; off += (bytes + 255) & ~(size_t)255; return p; };
  int*   cnt      = (int*)take((size_t)N_REL * N_NODES * 4);
  int*   rowstart = (int*)take((size_t)N_REL * N_NODES * 4);
  int*   cursor   = (int*)take((size_t)N_REL * N_NODES * 4);
  int*   srclist  = (int*)take((size_t)E * 4);
  float* tmp      = (float*)take((size_t)N_NODES * D_OUT * 4);
  float* acc1     = (float*)take((size_t)N_NODES * D_H * 4);
  float* acc2     = (float*)take((size_t)N_NODES * D_OUT * 4);
  float* pooled   = (float*)take((size_t)N_GRAPHS * 2 * D_OUT * 4);
  float* fc1o     = (float*)take((size_t)N_GRAPHS * 128 * 4);
  (void)ws_size; (void)n_in; (void)out_size;

  // ---- Build CSR by (relation, dst): counts, offsets, source lists ----
  {
    int n = N_REL * N_NODES;
    zero_i32_kernel<<<(n + 255) / 256, 256, 0, stream>>>(cnt, n);
    count_edges_kernel<<<(E + 255) / 256, 256, 0, stream>>>(dst, etype, cnt, E);
    exscan_kernel<<<1, 1024, 0, stream>>>(cnt, rowstart, cursor, n);
    fill_srclist_kernel<<<(E + 255) / 256, 256, 0, stream>>>(src, dst, etype, cursor, srclist, E);
  }

  // ---------------- Layer 1: [N,768] -> [N,512] ----------------
  gemm_wmma_f32_kernel<<<gemm_blocks(N_NODES, D_H), 256, 0, stream>>>(
      x, W1_root, b1, acc1, N_NODES, D_IN, D_H, 1);
  for (int r = 0; r < N_REL; ++r) {
    gemm_wmma_f32_kernel<<<gemm_blocks(N_NODES, D_H), 256, 0, stream>>>(
        x, W1_rel + (size_t)r * D_IN * D_H, nullptr, tmp, N_NODES, D_IN, D_H, 0);
    gather_mean_kernel<<<N_NODES, 256, 0, stream>>>(tmp, rowstart, cnt, srclist, acc1, D_H, r);
  }
  {
    long n = (long)N_NODES * D_H;
    relu_kernel<<<(int)((n + 255) / 256), 256, 0, stream>>>(acc1, n);
  }

  // ---------------- Layer 2: [N,512] -> [N,768] ----------------
  gemm_wmma_f32_kernel<<<gemm_blocks(N_NODES, D_OUT), 256, 0, stream>>>(
      acc1, W2_root, b2, acc2, N_NODES, D_H, D_OUT, 1);
  for (int r = 0; r < N_REL; ++r) {
    gemm_wmma_f32_kernel<<<gemm_blocks(N_NODES, D_OUT), 256, 0, stream>>>(
        acc1, W2_rel + (size_t)r * D_H * D_OUT, nullptr, tmp, N_NODES, D_H, D_OUT, 0);
    gather_mean_kernel<<<N_NODES, 256, 0, stream>>>(tmp, rowstart, cnt, srclist, acc2, D_OUT, r);
  }
  {
    long n = (long)N_NODES * D_OUT;
    relu_kernel<<<(int)((n + 255) / 256), 256, 0, stream>>>(acc2, n);
  }

  // ---------------- Readout ----------------
  {
    int n = N_GRAPHS * 2 * D_OUT;
    pool_kernel<<<(n + 255) / 256, 256, 0, stream>>>(acc2, pooled);
  }
  {
    int n = N_GRAPHS * 128;
    fc1_kernel<<<(n + 255) / 256, 256, 0, stream>>>(pooled, fc1_w, fc1_b, fc1o);
  }
  {
    int n = N_GRAPHS * 2;
    fc2_kernel<<<(n + 255) / 256, 256, 0, stream>>>(fc1o, fc2_w, fc2_b, out);
  }
}